// KrausRK4_10574209483121
// MI455X (gfx1250) — compile-verified
//
#include <hip/hip_runtime.h>

typedef __attribute__((ext_vector_type(2))) float v2f;
typedef __attribute__((ext_vector_type(8))) float v8f;

#define N 64
#define NN (N * N)

static __device__ __forceinline__ v8f zero8() { v8f z = {}; return z; }

// ---- fragment loaders (row-major source, stride N) -------------------------
// A fragment: rows m0..m0+15, cols k0..k0+3.
// Layout: lanes 0-15 -> M=lane, K = {k0,k0+1}; lanes 16-31 -> K = {k0+2,k0+3}.
static __device__ __forceinline__ v2f load_a(const float* M, int m0, int k0, int lane) {
    int r = m0 + (lane & 15);
    int c = k0 + ((lane >> 4) << 1);
    const float* p = M + r * N + c;
    v2f a; a.x = p[0]; a.y = p[1];
    return a;
}
// B fragment: rows k0..k0+3, cols n0..n0+15 of B (B stored row-major).
static __device__ __forceinline__ v2f load_b(const float* M, int k0, int n0, int lane) {
    int c = n0 + (lane & 15);
    int r = k0 + ((lane >> 4) << 1);
    v2f b; b.x = M[r * N + c]; b.y = M[(r + 1) * N + c];
    return b;
}
// B fragment where B = M^T (so B[k][n] = M[n][k]): contiguous pair load.
static __device__ __forceinline__ v2f load_bt(const float* M, int k0, int n0, int lane) {
    int r = n0 + (lane & 15);
    int c = k0 + ((lane >> 4) << 1);
    const float* p = M + r * N + c;
    v2f b; b.x = p[0]; b.y = p[1];
    return b;
}

// ---- C/D tile <-> row-major LDS --------------------------------------------
static __device__ __forceinline__ void store_tile(float* M, int m0, int n0, int lane, v8f c) {
    int r = m0 + ((lane >> 4) << 3);
    int col = n0 + (lane & 15);
#pragma unroll
    for (int j = 0; j < 8; ++j) M[(r + j) * N + col] = c[j];
}
static __device__ __forceinline__ v8f load_tile(const float* M, int m0, int n0, int lane) {
    int r = m0 + ((lane >> 4) << 3);
    int col = n0 + (lane & 15);
    v8f c;
#pragma unroll
    for (int j = 0; j < 8; ++j) c[j] = M[(r + j) * N + col];
    return c;
}

// ---- 64x64x64 matmul tiles via V_WMMA_F32_16X16X4_F32 ----------------------
static __device__ __forceinline__ v8f mm_nn(const float* A, const float* B,
                                            int m0, int n0, int lane, v8f c) {
#pragma unroll
    for (int k0 = 0; k0 < N; k0 += 4) {
        v2f a = load_a(A, m0, k0, lane);
        v2f b = load_b(B, k0, n0, lane);
        c = __builtin_amdgcn_wmma_f32_16x16x4_f32(false, a, false, b, (short)0, c,
                                                  false, false);
    }
    return c;
}
static __device__ __forceinline__ v8f mm_nt(const float* A, const float* B,
                                            int m0, int n0, int lane, v8f c) {
#pragma unroll
    for (int k0 = 0; k0 < N; k0 += 4) {
        v2f a = load_a(A, m0, k0, lane);
        v2f b = load_bt(B, k0, n0, lane);
        c = __builtin_amdgcn_wmma_f32_16x16x4_f32(false, a, false, b, (short)0, c,
                                                  false, false);
    }
    return c;
}

// tile of U * X * U^T (+acc).  X in LDS, U global/L2, sT LDS scratch.
static __device__ __forceinline__ v8f sandwich(const float* U, const float* X, float* sT,
                                               int m0, int n0, int lane, v8f acc) {
    v8f t = mm_nn(U, X, m0, n0, lane, zero8());
    __syncthreads();                 // prior readers of sT are done
    store_tile(sT, m0, n0, lane, t);
    __syncthreads();                 // sT (= U*X) fully visible
    return mm_nt(sT, U, m0, n0, lane, acc);
}

// tile of sum_k L_k * X * L_k^T.  Lbase = K row-major matrices in global.
static __device__ __forceinline__ v8f jump_sum(const float* Lbase, const float* X, float* sT,
                                               int m0, int n0, int lane) {
    v8f acc = zero8();
#pragma unroll 1
    for (int k = 0; k < 4; ++k) {
        const float* L = Lbase + k * NN;
        v8f t = mm_nn(L, X, m0, n0, lane, zero8());
        __syncthreads();             // prior readers of sT are done
        store_tile(sT, m0, n0, lane, t);
        __syncthreads();             // sT (= L*X) fully visible
        acc = mm_nt(sT, L, m0, n0, lane, acc);
    }
    return acc;
}

// ---------------------------------------------------------------------------
// Prep kernel: Um1 = U1 * inv(Um) via Gauss-Jordan in LDS (one block).
// ---------------------------------------------------------------------------
__global__ __launch_bounds__(256) void kraus_prep(const float* __restrict__ U1g,
                                                  const float* __restrict__ Umg,
                                                  float* __restrict__ Um1out) {
    __shared__ float M[N * 128];     // [Um | I]
    __shared__ float fac[N];
    const int tid = threadIdx.x;
    for (int i = tid; i < NN; i += 256) {
        int r = i >> 6, c = i & 63;
        M[r * 128 + c] = Umg[i];
        M[r * 128 + 64 + c] = (r == c) ? 1.0f : 0.0f;
    }
    __syncthreads();
    for (int c = 0; c < N; ++c) {
        float pivinv = 1.0f / M[c * 128 + c];
        __syncthreads();             // all read pivot before it is normalized
        if (tid < 128) M[c * 128 + tid] *= pivinv;
        __syncthreads();
        if (tid < N) fac[tid] = M[tid * 128 + c];
        __syncthreads();
        for (int i = tid; i < N * 128; i += 256) {
            int r = i >> 7, cc = i & 127;
            if (r != c) M[i] -= fac[r] * M[c * 128 + cc];
        }
        __syncthreads();
    }
    // Um1 = U1 * Minv
    for (int i = tid; i < NN; i += 256) {
        int r = i >> 6, cl = i & 63;
        float acc = 0.0f;
        for (int j = 0; j < N; ++j) acc += U1g[r * N + j] * M[j * 128 + 64 + cl];
        Um1out[i] = acc;
    }
}

// ---------------------------------------------------------------------------
// Main kernel: one block per batch element, 16 waves, one 16x16 tile each.
// ---------------------------------------------------------------------------
__global__ __launch_bounds__(512) void kraus_main(const float* __restrict__ rho0,
                                                  const float* __restrict__ U1g,
                                                  const float* __restrict__ Umg,
                                                  const float* __restrict__ Ls0,
                                                  const float* __restrict__ Lsmid,
                                                  const float* __restrict__ Ls1,
                                                  const float* __restrict__ dtp,
                                                  const float* __restrict__ Um1g,
                                                  float* __restrict__ out) {
    __shared__ float sR0[NN];        // rho0
    __shared__ float sT[NN];         // product scratch
    __shared__ float sX[NN];         // matmul staging
    __shared__ float sR2[NN];
    __shared__ float sR3[NN];
    __shared__ float sR4[NN];

    const int b = blockIdx.x;
    const int tid = threadIdx.x;
    const int lane = tid & 31;
    const int w = tid >> 5;
    const int m0 = (w >> 2) << 4;
    const int n0 = (w & 3) << 4;
    const float dt = dtp[0];

    // cooperative load of rho0 (4096 floats = 1024 float4)
    {
        const float4* src = (const float4*)(rho0 + (size_t)b * NN);
        float4* dst = (float4*)sR0;
        dst[tid] = src[tid];
        dst[tid + 512] = src[tid + 512];
    }
    __syncthreads();

    // P1 = jump_sum(Ls0, rho0)  (kept in registers; each wave needs only its tile)
    v8f p1 = jump_sum(Ls0, sR0, sT, m0, n0, lane);
    v8f r0t = load_tile(sR0, m0, n0, lane);

    // X1 = rho0 + 0.5*dt*P1
    {
        v8f x1;
#pragma unroll
        for (int j = 0; j < 8; ++j) x1[j] = r0t[j] + 0.5f * dt * p1[j];
        store_tile(sX, m0, n0, lane, x1);
    }
    __syncthreads();

    // rho2 = Um * X1 * Um^T
    v8f r2 = sandwich(Umg, sX, sT, m0, n0, lane, zero8());
    store_tile(sR2, m0, n0, lane, r2);

    // rho3 = Um*rho0*Um^T + 0.5*dt*jump_sum(Lsmid, rho2)
    v8f s3 = sandwich(Umg, sR0, sT, m0, n0, lane, zero8());  // internal barriers publish sR2
    v8f p2 = jump_sum(Lsmid, sR2, sT, m0, n0, lane);
    {
        v8f r3;
#pragma unroll
        for (int j = 0; j < 8; ++j) r3[j] = s3[j] + 0.5f * dt * p2[j];
        store_tile(sR3, m0, n0, lane, r3);
    }
    __syncthreads();

    // rho4 = U1*rho0*U1^T + Um1*(dt*jump_sum(Lsmid, rho3))*Um1^T
    v8f p3 = jump_sum(Lsmid, sR3, sT, m0, n0, lane);
    {
        v8f xt;
#pragma unroll
        for (int j = 0; j < 8; ++j) xt[j] = dt * p3[j];
        store_tile(sX, m0, n0, lane, xt);
    }
    __syncthreads();
    v8f r4 = sandwich(Um1g, sX, sT, m0, n0, lane, zero8());
    r4 = sandwich(U1g, sR0, sT, m0, n0, lane, r4);
    store_tile(sR4, m0, n0, lane, r4);
    __syncthreads();

    // out  = U1*(rho0 + dt/6*P1)*U1^T
    {
        v8f x6;
        const float c6 = dt * (1.0f / 6.0f);
#pragma unroll
        for (int j = 0; j < 8; ++j) x6[j] = r0t[j] + c6 * p1[j];
        store_tile(sX, m0, n0, lane, x6);
    }
    __syncthreads();
    v8f o = sandwich(U1g, sX, sT, m0, n0, lane, zero8());

    // rho23 = rho2 + rho3 -> sX (all waves past sandwich's first barrier => sX free)
    {
        v8f r2t = load_tile(sR2, m0, n0, lane);
        v8f r3t = load_tile(sR3, m0, n0, lane);
        v8f r23;
#pragma unroll
        for (int j = 0; j < 8; ++j) r23[j] = r2t[j] + r3t[j];
        store_tile(sX, m0, n0, lane, r23);
    }
    __syncthreads();

    //      + Um1*(dt/3 * jump_sum(Lsmid, rho23))*Um1^T
    v8f p23 = jump_sum(Lsmid, sX, sT, m0, n0, lane);
    {
        v8f x7;
        const float c3 = dt * (1.0f / 3.0f);
#pragma unroll
        for (int j = 0; j < 8; ++j) x7[j] = c3 * p23[j];
        store_tile(sX, m0, n0, lane, x7);
    }
    __syncthreads();
    o = sandwich(Um1g, sX, sT, m0, n0, lane, o);

    //      + dt/6 * jump_sum(Ls1, rho4)
    v8f p4 = jump_sum(Ls1, sR4, sT, m0, n0, lane);

    {
        const float c6 = dt * (1.0f / 6.0f);
        float* op = out + (size_t)b * NN;
        int r = m0 + ((lane >> 4) << 3);
        int col = n0 + (lane & 15);
#pragma unroll
        for (int j = 0; j < 8; ++j) op[(r + j) * N + col] = o[j] + c6 * p4[j];
    }
}

extern "C" void kernel_launch(void* const* d_in, const int* in_sizes, int n_in,
                              void* d_out, int out_size, void* d_ws, size_t ws_size,
                              hipStream_t stream) {
    const float* rho0  = (const float*)d_in[0];
    const float* U1    = (const float*)d_in[1];
    const float* Um    = (const float*)d_in[2];
    const float* Ls0   = (const float*)d_in[3];
    const float* Lsmid = (const float*)d_in[4];
    const float* Ls1   = (const float*)d_in[5];
    const float* dt    = (const float*)d_in[6];
    float* outp = (float*)d_out;
    float* Um1  = (float*)d_ws;    // 64*64 floats of scratch

    kraus_prep<<<1, 256, 0, stream>>>(U1, Um, Um1);
    kraus_main<<<4096, 512, 0, stream>>>(rho0, U1, Um, Ls0, Lsmid, Ls1, dt, Um1, outp);
}